// OSTL_63273458205168
// MI455X (gfx1250) — compile-verified
//
#include <hip/hip_runtime.h>

typedef __attribute__((ext_vector_type(2))) float        v2f;
typedef __attribute__((ext_vector_type(8))) float        v8f;
typedef __attribute__((ext_vector_type(4))) unsigned int v4u;
typedef __attribute__((ext_vector_type(8))) int          v8i;
typedef __attribute__((ext_vector_type(4))) int          v4i;

#define T_STEPS 512
#define IN_SZ   1024
#define OUT_SZ  1024

#define SIG_TAU 0.8807970779778823f   // sigmoid(2.0)
#define V_TH    1.0f

#define KCHUNK   32
#define BLK_M    128
#define BLK_N    64
#define XS_STRIDE 34    // 32 dwords + TDM pad (pad_interval=32dw, pad_amount=2dw)

#define HAVE_TDM __has_builtin(__builtin_amdgcn_tensor_load_to_lds)

// ---------------------------------------------------------------------------
// Build a 2-D TDM descriptor (CDNA5 ISA) and issue
// tensor_load_to_lds (6-arg toolchain variant: g0, g1, g2, g3, extra, cpol).
// data_size = 4 bytes. tile is tile_w x tile_h elements, rows separated by
// row_stride elements in global memory, packed into LDS starting at lds_addr
// (optionally with LDS padding via g1_flags).
// ---------------------------------------------------------------------------
#if HAVE_TDM
__device__ __forceinline__ void tdm_load_tile_2d(
    unsigned int lds_addr, const float* gptr,
    unsigned int tensor_rows, unsigned int tile_w, unsigned int tile_h,
    unsigned int row_stride_elts, unsigned int g1_flags)
{
  unsigned long long ga = (unsigned long long)(uintptr_t)gptr;
  v4u g0;
  g0.x = 1u;                                              // count=1, user desc
  g0.y = lds_addr;                                        // lds_addr[31:0]
  g0.z = (unsigned int)ga;                                // global_addr[31:0]
  g0.w = (unsigned int)((ga >> 32) & 0x01FFFFFFu) | (2u << 30); // addr[56:32] | type=2

  v8i g1;
  g1[0] = (int)((2u << 16) | g1_flags);                   // data_size=4B (+pad cfg)
  g1[1] = (int)((IN_SZ & 0xFFFFu) << 16);                 // tensor_dim0 lo16
  g1[2] = (int)((IN_SZ >> 16) | ((tensor_rows & 0xFFFFu) << 16)); // dim0 hi | dim1 lo
  g1[3] = (int)((tensor_rows >> 16) | (tile_w << 16));    // dim1 hi | tile_dim0
  g1[4] = (int)(tile_h);                                  // tile_dim1 (tile_dim2=0)
  g1[5] = (int)row_stride_elts;                           // tensor_dim0_stride lo32
  g1[6] = 0;
  g1[7] = 0;

  v4i g2 = {0, 0, 0, 0};
  v4i g3 = {0, 0, 0, 0};
  v8i g4 = {0, 0, 0, 0, 0, 0, 0, 0};                      // extra group (2D: unused)
  __builtin_amdgcn_tensor_load_to_lds(g0, g1, g2, g3, g4, 0);
}
#endif

// ---------------------------------------------------------------------------
// Phase 1: Y[T, OUT] = X[T, IN] @ W[IN, OUT], exact fp32 WMMA.
// Block: 8 waves, 128M x 64N tile; K staged through LDS in 32-wide chunks,
// double-buffered, fed by the Tensor Data Mover (wave 0 issues DMAs).
// ---------------------------------------------------------------------------
__global__ __launch_bounds__(256) void ostl_gemm_f32_wmma(
    const float* __restrict__ X,   // [T_STEPS, IN_SZ]
    const float* __restrict__ W,   // [IN_SZ, OUT_SZ]
    float* __restrict__ Y)         // [T_STEPS, OUT_SZ]
{
  __shared__ float Xs[2][BLK_M * XS_STRIDE];  // 128 rows x 34 (padded) floats
  __shared__ float Ws[2][KCHUNK * BLK_N];     // 32 rows x 64 floats, packed

  const int wave = threadIdx.x >> 5;
  const int lane = threadIdx.x & 31;

  // grid: 4 M-blocks x 16 N-blocks
  const int blk_m  = blockIdx.x >> 4;
  const int blk_n  = blockIdx.x & 15;
  const int m0     = blk_m * BLK_M;
  const int n_base = blk_n * BLK_N;

  const int half  = lane & 15;   // A row / B-&-D col within fragment
  const int khalf = lane >> 4;   // 0: K={0,1}, 1: K={2,3}

  v8f c0 = {}, c1 = {}, c2 = {}, c3 = {};

#if HAVE_TDM
  const unsigned int xs_lds[2] = { (unsigned int)(uintptr_t)&Xs[0][0],
                                   (unsigned int)(uintptr_t)&Xs[1][0] };
  const unsigned int ws_lds[2] = { (unsigned int)(uintptr_t)&Ws[0][0],
                                   (unsigned int)(uintptr_t)&Ws[1][0] };
  // X tile gets LDS padding: pad_enable(bit20), pad_interval=4 -> every 32
  // dwords (one row), pad_amount=1 -> 2 dwords  => LDS row stride 34 floats.
  const unsigned int xpad = (1u << 20) | (4u << 22) | (1u << 25);

  if (wave == 0) {  // prologue: chunk 0 into buffer 0
    tdm_load_tile_2d(xs_lds[0], X + (size_t)m0 * IN_SZ, T_STEPS,
                     KCHUNK, BLK_M, IN_SZ, xpad);
    tdm_load_tile_2d(ws_lds[0], W + n_base, IN_SZ,
                     BLK_N, KCHUNK, OUT_SZ, 0u);
  }

  int cur = 0;
  for (int kc = 0; kc < IN_SZ / KCHUNK; ++kc) {
    if (wave == 0) __builtin_amdgcn_s_wait_tensorcnt(0);
    __syncthreads();                              // buf[cur] ready for everyone

    if (wave == 0 && kc + 1 < IN_SZ / KCHUNK) {   // prefetch next chunk
      const int knext = (kc + 1) * KCHUNK;
      tdm_load_tile_2d(xs_lds[cur ^ 1], X + (size_t)m0 * IN_SZ + knext,
                       T_STEPS, KCHUNK, BLK_M, IN_SZ, xpad);
      tdm_load_tile_2d(ws_lds[cur ^ 1], W + (size_t)knext * OUT_SZ + n_base,
                       IN_SZ, BLK_N, KCHUNK, OUT_SZ, 0u);
    }
#else
  int cur = 0;
  for (int kc = 0; kc < IN_SZ / KCHUNK; ++kc) {
    {  // fallback: cooperative synchronous loads, single buffer
      const int kbase = kc * KCHUNK;
      for (int i = threadIdx.x; i < BLK_M * (KCHUNK / 4); i += 256) {
        const int r = i >> 3, q = i & 7;
        const float4 v = *(const float4*)(X + (size_t)(m0 + r) * IN_SZ + kbase + q * 4);
        float* dst = &Xs[0][r * XS_STRIDE + q * 4];
        dst[0] = v.x; dst[1] = v.y; dst[2] = v.z; dst[3] = v.w;
      }
      for (int i = threadIdx.x; i < KCHUNK * (BLK_N / 4); i += 256) {
        const int r = i >> 4, q = i & 15;
        *(float4*)(&Ws[0][r * BLK_N + q * 4]) =
            *(const float4*)(W + (size_t)(kbase + r) * OUT_SZ + n_base + q * 4);
      }
    }
    __syncthreads();
#endif

    // ---- compute 8 K-substeps of 4 from LDS ----
    const float* xw = &Xs[cur][(wave * 16 + half) * XS_STRIDE + khalf * 2];
#pragma unroll
    for (int k4 = 0; k4 < KCHUNK / 4; ++k4) {
      v2f a;
      a.x = xw[k4 * 4];
      a.y = xw[k4 * 4 + 1];

      const int krow = k4 * 4 + khalf * 2;
      const float* wr0 = &Ws[cur][krow * BLK_N + half];
      const float* wr1 = wr0 + BLK_N;
      v2f b0, b1, b2, b3;
      b0.x = wr0[0];  b0.y = wr1[0];
      b1.x = wr0[16]; b1.y = wr1[16];
      b2.x = wr0[32]; b2.y = wr1[32];
      b3.x = wr0[48]; b3.y = wr1[48];

      c0 = __builtin_amdgcn_wmma_f32_16x16x4_f32(false, a, false, b0, (short)0, c0, false, false);
      c1 = __builtin_amdgcn_wmma_f32_16x16x4_f32(false, a, false, b1, (short)0, c1, false, false);
      c2 = __builtin_amdgcn_wmma_f32_16x16x4_f32(false, a, false, b2, (short)0, c2, false, false);
      c3 = __builtin_amdgcn_wmma_f32_16x16x4_f32(false, a, false, b3, (short)0, c3, false, false);
    }

    __syncthreads();   // everyone done reading buf[cur] before it is reloaded
#if HAVE_TDM
    cur ^= 1;
#endif
  }

  // ---- store D tiles: VGPR r -> M = r + 8*khalf, N = half ----
  const int srow = m0 + wave * 16 + khalf * 8;
  float* yout = Y + (size_t)srow * OUT_SZ + n_base + half;
#pragma unroll
  for (int r = 0; r < 8; ++r) {
    yout[r * OUT_SZ + 0]  = c0[r];
    yout[r * OUT_SZ + 16] = c1[r];
    yout[r * OUT_SZ + 32] = c2[r];
    yout[r * OUT_SZ + 48] = c3[r];
  }
}

// ---------------------------------------------------------------------------
// Phase 2: per-neuron LIF scan (leak + threshold + soft reset), coalesced.
// ---------------------------------------------------------------------------
__global__ __launch_bounds__(256) void ostl_scan(
    const float* __restrict__ Y,
    const float* __restrict__ u0,
    float* __restrict__ spikes,
    float* __restrict__ u_final)
{
  const int n = blockIdx.x * blockDim.x + threadIdx.x;
  if (n >= OUT_SZ) return;

  float u = u0[n];
#pragma unroll 8
  for (int t = 0; t < T_STEPS; ++t) {
    const float u_new = SIG_TAU * u + Y[t * OUT_SZ + n];
    const float s = (u_new > V_TH) ? 1.0f : 0.0f;
    spikes[t * OUT_SZ + n] = s;
    u = u_new - s * V_TH;
  }
  u_final[n] = u;
}

// ---------------------------------------------------------------------------
// d_in: [0]=x_seq, [1]=kernel, [2]=u0, [3]=E0 (unused in primal)
// d_out: spikes (512*1024) ++ u_final (1024), f32
// d_ws:  Y scratch (2 MB)
// ---------------------------------------------------------------------------
extern "C" void kernel_launch(void* const* d_in, const int* in_sizes, int n_in,
                              void* d_out, int out_size, void* d_ws, size_t ws_size,
                              hipStream_t stream) {
  (void)in_sizes; (void)n_in; (void)out_size; (void)ws_size;

  const float* x_seq  = (const float*)d_in[0];
  const float* weight = (const float*)d_in[1];
  const float* u0     = (const float*)d_in[2];

  float* spikes  = (float*)d_out;
  float* u_final = spikes + (size_t)T_STEPS * OUT_SZ;
  float* Y       = (float*)d_ws;

  ostl_gemm_f32_wmma<<<64, 256, 0, stream>>>(x_seq, weight, Y);
  ostl_scan<<<(OUT_SZ + 255) / 256, 256, 0, stream>>>(Y, u0, spikes, u_final);
}